// AttentionCropLayer_27487790695101
// MI455X (gfx1250) — compile-verified
//
#include <hip/hip_runtime.h>
#include <hip/hip_bf16.h>
#include <stdint.h>

#define IN 108
#define PLANE (IN * IN)

typedef uint32_t u32x4 __attribute__((ext_vector_type(4)));
typedef int      i32x4 __attribute__((ext_vector_type(4)));
typedef int      i32x8 __attribute__((ext_vector_type(8)));
typedef float    f32x4 __attribute__((ext_vector_type(4)));

#if defined(__HIP_DEVICE_COMPILE__) && __has_builtin(__builtin_amdgcn_tensor_load_to_lds)
#define USE_TDM 1
#else
#define USE_TDM 0
#endif

__device__ __forceinline__ float sig10(float v) {
    return 1.0f / (1.0f + __expf(-10.0f * v));
}

__global__ __launch_bounds__(256) void attention_crop_kernel(
    const float* __restrict__ images, const float* __restrict__ locs,
    float* __restrict__ out) {
    __shared__ float s[PLANE];           // one 108x108 fp32 plane (46656 B)
    __shared__ f32x4 rpack[IN];          // (rw0, rw1, r0*108 bits, r1*108 bits)
    __shared__ f32x4 cpack[IN];          // (cw0, cw1, c0 bits,     c1 bits)

    const int tid      = threadIdx.x;
    const int plane_id = blockIdx.x;     // n*3 + ch
    const int n        = plane_id / 3;
    const float* __restrict__ src = images + (size_t)plane_id * PLANE;

#if USE_TDM
    // ---- TDM DMA: global plane -> LDS, issued once by wave 0, overlaps weight setup ----
    if (tid < 32) {
        uint64_t ga       = (uint64_t)(uintptr_t)src;
        uint32_t lds_addr = (uint32_t)(uintptr_t)(void*)s;   // low 32 bits of flat = LDS byte addr
        u32x4 g0;
        g0[0] = 1u;                                          // count=1 (valid user descriptor)
        g0[1] = lds_addr;                                    // lds_addr [63:32]
        g0[2] = (uint32_t)ga;                                // global_addr [95:64]
        g0[3] = (uint32_t)((ga >> 32) & 0x01FFFFFFu)         // global_addr [120:96]
              | (2u << 30);                                  // type=2 ("image") [127:126]
        i32x8 g1;
        g1[0] = 0x00020000;                                  // data_size=2 -> 4 bytes; wg_mask=0
        g1[1] = (int)(IN << 16);                             // tensor_dim0 lo16 = 108
        g1[2] = (int)(IN << 16);                             // tensor_dim0 hi=0 | tensor_dim1 lo16 = 108
        g1[3] = (int)(IN << 16);                             // tensor_dim1 hi=0 | tile_dim0 = 108
        g1[4] = IN;                                          // tile_dim1 = 108 | tile_dim2 = 0
        g1[5] = IN;                                          // tensor_dim0_stride lo32 = 108 elements
        g1[6] = 0;
        g1[7] = 0;
        i32x4 gz = {0, 0, 0, 0};
#if __clang_major__ >= 23
        i32x8 gz8 = {0, 0, 0, 0, 0, 0, 0, 0};
        __builtin_amdgcn_tensor_load_to_lds(g0, g1, gz, gz, gz8, 0);
#else
        __builtin_amdgcn_tensor_load_to_lds(g0, g1, gz, gz, 0);
#endif
    }
#else
    // Fallback: cooperative b128 copy global -> LDS
    {
        const f32x4* __restrict__ s4 = (const f32x4*)src;
        f32x4* d4 = (f32x4*)s;
        for (int i = tid; i < PLANE / 4; i += 256) d4[i] = s4[i];
    }
#endif

    // ---- soft-box scalars (uniform per block) ----
    const float l0 = locs[n * 3 + 0];
    const float l1 = locs[n * 3 + 1];
    const float l2 = locs[n * 3 + 2];
    const float tx = 54.0f + truncf(l0 * 27.0f + 0.5f);
    const float ty = 54.0f + truncf(l1 * 27.0f + 0.5f);
    const float tl = 21.0f + truncf(l2 * 7.0f + 0.5f);
    const float w_off = fmaxf(tx - tl, 0.0f);
    const float h_off = fmaxf(ty - tl, 0.0f);
    const float w_end = fminf(tx + tl, 108.0f);
    const float h_end = fminf(ty + tl, 108.0f);

    // ---- separable bilinear+mask weight tables (mask folded into taps) ----
    if (tid < IN) {
        const int j  = tid;
        float sr = w_off + (float)j * (w_end - w_off - 1.0f) / 107.0f;
        int   r0 = (int)fminf(fmaxf(floorf(sr), 0.0f), 107.0f);
        int   r1 = r0 + 1; if (r1 > 107) r1 = 107;
        float fr = sr - (float)r0;
        float m0 = sig10((float)r0 - w_off) - sig10((float)r0 - w_end);
        float m1 = sig10((float)r1 - w_off) - sig10((float)r1 - w_end);
        f32x4 rp;
        rp.x = (1.0f - fr) * m0;
        rp.y = fr * m1;
        rp.z = __int_as_float(r0 * IN);
        rp.w = __int_as_float(r1 * IN);
        rpack[j] = rp;
    } else if (tid >= 128 && tid < 128 + IN) {
        const int k  = tid - 128;
        float sc = h_off + (float)k * (h_end - h_off - 1.0f) / 107.0f;
        int   c0 = (int)fminf(fmaxf(floorf(sc), 0.0f), 107.0f);
        int   c1 = c0 + 1; if (c1 > 107) c1 = 107;
        float fc = sc - (float)c0;
        float m0 = sig10((float)c0 - h_off) - sig10((float)c0 - h_end);
        float m1 = sig10((float)c1 - h_off) - sig10((float)c1 - h_end);
        f32x4 cp;
        cp.x = (1.0f - fc) * m0;
        cp.y = fc * m1;
        cp.z = __int_as_float(c0);
        cp.w = __int_as_float(c1);
        cpack[k] = cp;
    }

#if USE_TDM
    if (tid < 32) __builtin_amdgcn_s_wait_tensorcnt(0);   // TDM data landed in LDS
#endif
    __syncthreads();

    // ---- 4-tap gather from LDS, 4 pixels per thread, b128 streaming stores ----
    // 108 % 4 == 0 -> each float4 output group lies entirely within one row.
    f32x4* __restrict__ dst4 = (f32x4*)(out + (size_t)plane_id * PLANE);
    for (int q = tid; q < PLANE / 4; q += 256) {
        const int j  = q / (IN / 4);            // row (27 float4 groups per row)
        const int k0 = (q - j * (IN / 4)) * 4;  // starting column
        const f32x4 rp = rpack[j];
        const int r0 = __float_as_int(rp.z);
        const int r1 = __float_as_int(rp.w);
        f32x4 ov;
#pragma unroll
        for (int t = 0; t < 4; ++t) {
            const f32x4 cp = cpack[k0 + t];
            const int c0 = __float_as_int(cp.z);
            const int c1 = __float_as_int(cp.w);
            const float row0 = s[r0 + c0] * rp.x + s[r1 + c0] * rp.y;
            const float row1 = s[r0 + c1] * rp.x + s[r1 + c1] * rp.y;
            ov[t] = row0 * cp.x + row1 * cp.y;
        }
        __builtin_nontemporal_store(ov, &dst4[q]);   // streamed out, never re-read
    }
}

extern "C" void kernel_launch(void* const* d_in, const int* in_sizes, int n_in,
                              void* d_out, int out_size, void* d_ws, size_t ws_size,
                              hipStream_t stream) {
    const float* images = (const float*)d_in[0];   // [1024,3,108,108] fp32
    const float* locs   = (const float*)d_in[1];   // [1024,3] fp32
    float* out          = (float*)d_out;           // [1024,3,108,108] fp32
    const int planes = in_sizes[0] / PLANE;        // 1024*3 = 3072
    attention_crop_kernel<<<planes, 256, 0, stream>>>(images, locs, out);
}